// Wide_75728863363405
// MI455X (gfx1250) — compile-verified
//
#include <hip/hip_runtime.h>
#include <hip/hip_bf16.h>

typedef __attribute__((ext_vector_type(2))) float v2f;
typedef __attribute__((ext_vector_type(8))) float v8f;

#define BATCH       32768
#define DENSE_DIM   16
#define OUT_DIM     128
#define FEATURE_DIM 3415

// Feature-column layout inside W's second axis:
//  [0,16)      dense
//  [16,40)     one_hot(s0, 24)
//  [40,47)     one_hot(s1, 7)
//  [47,147)    one_hot(s2, 100)
//  [147,315)   one_hot(7*s0 + s1, 168)
//  [315,2715)  one_hot(100*s0 + s2, 2400)
//  [2715,3415) one_hot(100*s1 + s2, 700)
#define COL_S0   16
#define COL_S1   40
#define COL_S2   47
#define COL_S01  147
#define COL_S02  315
#define COL_S12  2715

// W [128, 3415] -> Wt [3415, 128]. Writes fully coalesced (128 consecutive
// floats per block); strided reads are served by L2 (W is only 1.75 MB).
__global__ __launch_bounds__(OUT_DIM)
void Wide_transpose_kernel(const float* __restrict__ W, float* __restrict__ Wt) {
    const int c = blockIdx.x;    // feature column 0..3414
    const int o = threadIdx.x;   // output 0..127
    Wt[c * OUT_DIM + o] = W[o * FEATURE_DIM + c];
}

// One block = 16 batch rows x 128 outputs. 8 waves; wave w computes the
// 16x16 f32 tile for outputs [16w, 16w+16) using 4x V_WMMA_F32_16X16X4_F32
// over the 16 dense features, then adds bias + 6 gathered W columns (the
// one-hot / cross terms) and applies ReLU. USE_WT selects the transposed
// (coalesced) weight layout at compile time — no per-load branching.
template <bool USE_WT>
__global__ __launch_bounds__(256)
void Wide_75728863363405_kernel(const float* __restrict__ dense,
                                const int*   __restrict__ sparse,
                                const float* __restrict__ W,
                                const float* __restrict__ Wt,
                                const float* __restrict__ bias,
                                float*       __restrict__ out) {
    __shared__ int cols[16][6];

    const int tid  = threadIdx.x;
    const int lane = tid & 31;
    const int wv   = tid >> 5;            // 0..7 : which 16-wide output tile
    const int row0 = blockIdx.x * 16;     // first batch row of this block

    // Decode the 6 fused gather columns for each of the 16 rows into LDS.
    if (tid < 16) {
        const int r  = row0 + tid;
        const int s0 = sparse[r * 3 + 0];
        const int s1 = sparse[r * 3 + 1];
        const int s2 = sparse[r * 3 + 2];
        cols[tid][0] = COL_S0  + s0;
        cols[tid][1] = COL_S1  + s1;
        cols[tid][2] = COL_S2  + s2;
        cols[tid][3] = COL_S01 + s0 * 7   + s1;
        cols[tid][4] = COL_S02 + s0 * 100 + s2;
        cols[tid][5] = COL_S12 + s1 * 100 + s2;
    }
    __syncthreads();

    const int half = lane >> 4;           // lane group 0 / 1
    const int l16  = lane & 15;
    const int o    = wv * 16 + l16;       // output column for B/C/D fragments

    // ---- load all fragments up front (clustered b64/b32 loads, one wait)
    v2f a[4], bf[4];
#pragma unroll
    for (int i = 0; i < 4; ++i) {
        const int kb = i * 4 + half * 2;
        // A fragment (16x4): lane holds M = l16, K = kb, kb+1 (8B aligned pair)
        a[i] = *(const v2f*)&dense[(row0 + l16) * DENSE_DIM + kb];
        // B fragment (4x16): lane holds N = l16, K = kb, kb+1 ; B[k][n] = W[n][k]
        if (USE_WT) {
            bf[i].x = Wt[kb       * OUT_DIM + o];
            bf[i].y = Wt[(kb + 1) * OUT_DIM + o];
        } else {
            bf[i] = *(const v2f*)&W[o * FEATURE_DIM + kb];
        }
    }

    // ---- dense[16x16] @ Wdense[16x16]^T via 4 chained f32 WMMAs (K=4 each)
    v8f acc = {};
#pragma unroll
    for (int i = 0; i < 4; ++i) {
        acc = __builtin_amdgcn_wmma_f32_16x16x4_f32(
            /*neg_a=*/false, a[i], /*neg_b=*/false, bf[i],
            /*c_mod=*/(short)0, acc, /*reuse_a=*/false, /*reuse_b=*/false);
    }

    // ---- epilogue: bias + 6 one-hot/cross gathers + ReLU, coalesced stores
    const float bo = bias[o];
#pragma unroll
    for (int v = 0; v < 8; ++v) {
        const int m = half * 8 + v;       // row within tile held by acc[v]
        float g = bo;
#pragma unroll
        for (int j = 0; j < 6; ++j) {
            const int c = cols[m][j];
            g += USE_WT ? Wt[c * OUT_DIM + o] : W[o * FEATURE_DIM + c];
        }
        const float r = acc[v] + g;
        out[(row0 + m) * OUT_DIM + o] = r > 0.f ? r : 0.f;
    }
}

extern "C" void kernel_launch(void* const* d_in, const int* in_sizes, int n_in,
                              void* d_out, int out_size, void* d_ws, size_t ws_size,
                              hipStream_t stream) {
    const float* dense  = (const float*)d_in[0];
    const int*   sparse = (const int*)d_in[1];
    const float* W      = (const float*)d_in[2];
    const float* bias   = (const float*)d_in[3];
    float*       out    = (float*)d_out;
    float*       Wt     = (float*)d_ws;

    const size_t wt_bytes = (size_t)FEATURE_DIM * OUT_DIM * sizeof(float);

    if (ws_size >= wt_bytes) {
        Wide_transpose_kernel<<<FEATURE_DIM, OUT_DIM, 0, stream>>>(W, Wt);
        Wide_75728863363405_kernel<true><<<BATCH / 16, 256, 0, stream>>>(
            dense, sparse, W, Wt, bias, out);
    } else {
        Wide_75728863363405_kernel<false><<<BATCH / 16, 256, 0, stream>>>(
            dense, sparse, W, Wt, bias, out);
    }
}